// StandaloneGRU_27307402068614
// MI455X (gfx1250) — compile-verified
//
#include <hip/hip_runtime.h>
#include <stdint.h>

typedef __attribute__((ext_vector_type(2))) float v2f;
typedef __attribute__((ext_vector_type(8))) float v8f;

#define B_  4096
#define T_  512
#define N_  36
#define H_  16

#define XPAD 38   // even (b64-aligned), 38*r mod 64 distinct for r<16 -> conflict-free frag reads
#define HPAD 18   // even, 18*r mod 64 distinct for r<16
#define OPAD 38

#define XTILE (16 * XPAD)
#define HTILE (16 * HPAD)
#define OTILE (16 * OPAD)
#define STILE (16 * 4)

static __device__ __forceinline__ v8f wmma4(v2f a, v2f b, v8f c) {
    // D = A(16x4, f32) * B(4x16, f32) + C(16x16, f32)
    return __builtin_amdgcn_wmma_f32_16x16x4_f32(false, a, false, b, (short)0, c, false, false);
}

static __device__ __forceinline__ void wait_ds() {
    asm volatile("s_wait_dscnt 0" ::: "memory");
}
// wait until at most the 9 just-issued (next-tile) async loads remain outstanding
static __device__ __forceinline__ void wait_async_prev_tile() {
    asm volatile("s_wait_asynccnt 0x9" ::: "memory");
}

// Async-copy 18 contiguous floats (this lane's slice) global -> LDS.
// IOFFSET is added to BOTH the LDS and global addresses (ISA 10.7/15.18), so one
// base pair + immediate offsets covers the whole 72-byte slice.
static __device__ __forceinline__ void async_copy_slice(uint32_t lds_addr, const float* g) {
    asm volatile(
        "global_load_async_to_lds_b64 %0, %1, off\n\t"
        "global_load_async_to_lds_b64 %0, %1, off offset:8\n\t"
        "global_load_async_to_lds_b64 %0, %1, off offset:16\n\t"
        "global_load_async_to_lds_b64 %0, %1, off offset:24\n\t"
        "global_load_async_to_lds_b64 %0, %1, off offset:32\n\t"
        "global_load_async_to_lds_b64 %0, %1, off offset:40\n\t"
        "global_load_async_to_lds_b64 %0, %1, off offset:48\n\t"
        "global_load_async_to_lds_b64 %0, %1, off offset:56\n\t"
        "global_load_async_to_lds_b64 %0, %1, off offset:64"
        :: "v"(lds_addr), "v"(g) : "memory");
}

#if __has_builtin(__builtin_amdgcn_tanhf)
static __device__ __forceinline__ float fast_tanh(float x) { return __builtin_amdgcn_tanhf(x); }
#else
static __device__ __forceinline__ float fast_tanh(float x) { return tanhf(x); }
#endif
static __device__ __forceinline__ float fast_sigmoid(float x) {
    return 0.5f + 0.5f * fast_tanh(0.5f * x); // one v_tanh_f32 trans op
}

__global__ void __launch_bounds__(32)
gru_fused_wmma(const float* __restrict__ x,     // [B,T,N]
               const float* __restrict__ h0,    // [B,H]
               const float* __restrict__ W_ih,  // [3H,N]
               const float* __restrict__ W_hh,  // [3H,H]
               const float* __restrict__ b_ih,  // [3H]
               const float* __restrict__ b_hh,  // [3H]
               const float* __restrict__ W_s,   // [3,H]
               const float* __restrict__ b_s,   // [3]
               const float* __restrict__ W_o,   // [N,H]
               const float* __restrict__ b_o,   // [N]
               float* __restrict__ out_state,   // [B,T,3]
               float* __restrict__ out_ori)     // [B,T,N]
{
    __shared__ float lds[2 * XTILE + HTILE + OTILE + STILE];
    float* xs0 = lds;                 // x tile double buffer 0
    float* xs1 = lds + XTILE;         // x tile double buffer 1
    float* hs  = lds + 2 * XTILE;     // h tile [16][HPAD]
    float* os  = hs + HTILE;          // ori out tile [16][OPAD]
    float* ss  = os + OTILE;          // state out [16][4]

    const int lane = threadIdx.x & 31;
    const int b0   = blockIdx.x * 16; // one 16-row batch tile per wave/block
    const int j    = lane & 15;
    const int half = lane >> 4;

    // ---------------- weight B-fragments (col = 16n+j, VGPR d <-> K = kbase+2*half+d) ----------
    v2f bih[3][9], bhh[3][4], bwo[3][4];
    #pragma unroll
    for (int n = 0; n < 3; ++n) {
        const int col = 16 * n + j;
        #pragma unroll
        for (int k = 0; k < 9; ++k)
            bih[n][k] = *(const v2f*)(W_ih + col * N_ + 4 * k + 2 * half);
        #pragma unroll
        for (int k = 0; k < 4; ++k)
            bhh[n][k] = *(const v2f*)(W_hh + col * H_ + 4 * k + 2 * half);
        #pragma unroll
        for (int k = 0; k < 4; ++k) {
            const int kk = 4 * k + 2 * half;
            v2f w;
            if (col < 3)       w = *(const v2f*)(W_s + col * H_ + kk);
            else if (col < 39) w = *(const v2f*)(W_o + (col - 3) * H_ + kk);
            else               w = (v2f){0.0f, 0.0f};
            bwo[n][k] = w;
        }
    }

    // ---------------- bias fragments: biases folded into WMMAs via a one-hot K-slice ------------
    // A one-hot frag: A[r][kbase] = 1 for all r  (lane half==0, d==0)
    const float oh = (half == 0) ? 1.0f : 0.0f;
    const v2f onehot = {oh, 0.0f};
    const v2f brzf0  = {(half == 0) ? (b_ih[j]      + b_hh[j])      : 0.0f, 0.0f};
    const v2f brzf1  = {(half == 0) ? (b_ih[16 + j] + b_hh[16 + j]) : 0.0f, 0.0f};
    const v2f binf   = {(half == 0) ? b_ih[32 + j] : 0.0f, 0.0f};
    const v2f bhnf   = {(half == 0) ? b_hh[32 + j] : 0.0f, 0.0f};
    v2f boutf[3];
    #pragma unroll
    for (int n = 0; n < 3; ++n) {
        const int col = 16 * n + j;
        const float bv = (col < 3) ? b_s[col] : ((col < 39) ? b_o[col - 3] : 0.0f);
        boutf[n] = (v2f){(half == 0) ? bv : 0.0f, 0.0f};
    }

    // ---------------- h state: D-layout regs + A-layout frags via LDS ---------------------------
    v8f hreg; // element v: h[m = v + 8*half][j]
    #pragma unroll
    for (int v = 0; v < 8; ++v)
        hreg[v] = h0[(size_t)(b0 + v + 8 * half) * H_ + j];
    #pragma unroll
    for (int v = 0; v < 8; ++v)
        hs[(v + 8 * half) * HPAD + j] = hreg[v];
    wait_ds();
    v2f hA[4]; // A frag: row = j, K = 4k + 2*half + d
    #pragma unroll
    for (int k = 0; k < 4; ++k)
        hA[k] = *(const v2f*)(hs + j * HPAD + 4 * k + 2 * half);

    // ---------------- async x prefetch: this lane's 18-float slice ------------------------------
    const float* xrow = x + (size_t)(b0 + j) * T_ * N_ + half * 18;
    const uint32_t ldsx[2] = {
        (uint32_t)(uintptr_t)(xs0 + j * XPAD + half * 18),
        (uint32_t)(uintptr_t)(xs1 + j * XPAD + half * 18)
    };
    async_copy_slice(ldsx[0], xrow); // prefetch tile t = 0

    const v8f zc = {}; // zero C -> inline-0 SRC2 in first WMMA of each chain

    for (int t = 0; t < T_; ++t) {
        const int cb = t & 1, nb = (t + 1) & 1;

        // prefetch next tile (wrap at the end so the wait semantics stay uniform)
        wait_ds(); // close LDS WAR window on the buffer we are about to overwrite
        const int tn = (t + 1 < T_) ? (t + 1) : 0;
        async_copy_slice(ldsx[nb], xrow + (size_t)tn * N_);

        // ---- hidden GEMM: gh = h @ W_hh^T (zero-init C inline) ----
        v8f acc_r  = wmma4(hA[0], bhh[0][0], zc);
        v8f acc_z  = wmma4(hA[0], bhh[1][0], zc);
        v8f acc_hn = wmma4(hA[0], bhh[2][0], zc);
        #pragma unroll
        for (int k = 1; k < 4; ++k) {
            acc_r  = wmma4(hA[k], bhh[0][k], acc_r);
            acc_z  = wmma4(hA[k], bhh[1][k], acc_z);
            acc_hn = wmma4(hA[k], bhh[2][k], acc_hn);
        }
        acc_hn = wmma4(onehot, bhnf, acc_hn);   // + b_hh[n-gate]
        v8f acc_in = wmma4(onehot, binf, zc);   // b_ih[n-gate]

        wait_async_prev_tile(); // current tile resident in LDS

        // ---- input GEMM: gi = x_t @ W_ih^T (K = 36 = 9*4) ----
        const float* xc = (cb == 0) ? xs0 : xs1;
        #pragma unroll
        for (int k = 0; k < 9; ++k) {
            v2f aX = *(const v2f*)(xc + j * XPAD + 4 * k + 2 * half);
            acc_r  = wmma4(aX, bih[0][k], acc_r);
            acc_z  = wmma4(aX, bih[1][k], acc_z);
            acc_in = wmma4(aX, bih[2][k], acc_in);
        }
        acc_r = wmma4(onehot, brzf0, acc_r);    // + (b_ih + b_hh)[r]
        acc_z = wmma4(onehot, brzf1, acc_z);    // + (b_ih + b_hh)[z]

        // ---- gates + state update (per-lane VALU, v_tanh trans ops) ----
        #pragma unroll
        for (int v = 0; v < 8; ++v) {
            float r  = fast_sigmoid(acc_r[v]);
            float z  = fast_sigmoid(acc_z[v]);
            float nn = fast_tanh(acc_in[v] + r * acc_hn[v]);
            hreg[v]  = (1.0f - z) * nn + z * hreg[v];
        }

        // ---- transpose h_new D-layout -> A-layout through LDS ----
        #pragma unroll
        for (int v = 0; v < 8; ++v)
            hs[(v + 8 * half) * HPAD + j] = hreg[v];
        wait_ds();
        #pragma unroll
        for (int k = 0; k < 4; ++k)
            hA[k] = *(const v2f*)(hs + j * HPAD + 4 * k + 2 * half);

        // ---- output GEMM: [state|ori] = h_new @ [W_s;W_o]^T (48 padded cols) ----
        v8f oacc[3];
        #pragma unroll
        for (int n = 0; n < 3; ++n) {
            oacc[n] = wmma4(onehot, boutf[n], zc); // bias init
            #pragma unroll
            for (int k = 0; k < 4; ++k)
                oacc[n] = wmma4(hA[k], bwo[n][k], oacc[n]);
        }

        // ---- re-layout outputs through LDS for contiguous global stores ----
        #pragma unroll
        for (int n = 0; n < 3; ++n) {
            #pragma unroll
            for (int v = 0; v < 8; ++v) {
                const int c = 16 * n + j;
                const int m = v + 8 * half;
                if (c < 3)       ss[m * 4 + c]          = oacc[n][v];
                else if (c < 39) os[m * OPAD + (c - 3)] = oacc[n][v];
            }
        }
        wait_ds();

        const size_t rowbase = (size_t)(b0 + j) * T_ + t;
        float* op = out_ori + rowbase * (size_t)N_ + half * 18;
        #pragma unroll
        for (int i = 0; i < 9; ++i)
            *(v2f*)(op + 2 * i) = *(const v2f*)(os + j * OPAD + half * 18 + 2 * i);
        if (half == 0) {
            float* sp = out_state + rowbase * 3;
            sp[0] = ss[j * 4 + 0];
            sp[1] = ss[j * 4 + 1];
            sp[2] = ss[j * 4 + 2];
        }
    }
}

extern "C" void kernel_launch(void* const* d_in, const int* in_sizes, int n_in,
                              void* d_out, int out_size, void* d_ws, size_t ws_size,
                              hipStream_t stream) {
    (void)in_sizes; (void)n_in; (void)out_size; (void)d_ws; (void)ws_size;
    const float* x    = (const float*)d_in[0];
    const float* h0   = (const float*)d_in[1];
    const float* W_ih = (const float*)d_in[2];
    const float* W_hh = (const float*)d_in[3];
    const float* b_ih = (const float*)d_in[4];
    const float* b_hh = (const float*)d_in[5];
    const float* W_s  = (const float*)d_in[6];
    const float* b_s  = (const float*)d_in[7];
    const float* W_o  = (const float*)d_in[8];
    const float* b_o  = (const float*)d_in[9];

    float* out_state = (float*)d_out;                       // [B,T,3]
    float* out_ori   = (float*)d_out + (size_t)B_ * T_ * 3; // [B,T,N]

    dim3 grid(B_ / 16);   // 256 blocks: one wave (one 16-row batch tile) each
    dim3 block(32);
    gru_fused_wmma<<<grid, block, 0, stream>>>(x, h0, W_ih, W_hh, b_ih, b_hh,
                                               W_s, b_s, W_o, b_o,
                                               out_state, out_ori);
}